// FNet_30193620091461
// MI455X (gfx1250) — compile-verified
//
#include <hip/hip_runtime.h>
#include <hip/hip_bf16.h>
#include <stdint.h>

// Problem geometry (fixed by the reference)
#define NLOC    8192            // feature dim / number of locations
#define TOKENS  (512 * 64)      // 32768 gathered rows
#define JTILE   1024            // j-span per block: 256 threads * 4 floats
#define TB      8               // tokens per block (share the bias tile / L2 slab)

typedef __attribute__((ext_vector_type(4))) int v4i;
typedef __attribute__((address_space(1))) v4i* global_v4i_p;  // AS1 (prints as __device__)
typedef __attribute__((address_space(3))) v4i* lds_v4i_p;     // AS3

// ---- CDNA5 async-copy feature detection -----------------------------------
#if defined(__has_builtin)
# if __has_builtin(__builtin_amdgcn_global_load_async_to_lds_b128)
#  define HAS_ASYNC_B128 1
# endif
# if __has_builtin(__builtin_amdgcn_s_wait_asynccnt)
#  define HAS_WAIT_ASYNC 1
# endif
#endif

__global__ __launch_bounds__(256) void fnet_gather_bias_kernel(
    const int*   __restrict__ widx,   // [TOKENS] gather indices
    const float* __restrict__ Wm,     // [NLOC, NLOC] row-major
    const float* __restrict__ bias,   // [NLOC]
    float*       __restrict__ out)    // [TOKENS, NLOC]
{
    const int tid = threadIdx.x;
    const int j0  = blockIdx.y * JTILE;       // j-slab (slow-varying -> L2 slab reuse)
    const int t0  = blockIdx.x * TB;          // token tile (fast-varying)
    const int j   = j0 + tid * 4;             // this thread's 4 consecutive features

    __shared__ float sb[JTILE];

    // ---- Stage bias tile into LDS with the CDNA5 async path (ASYNCcnt) ----
    {
        const float* src = bias + j;
#if HAS_ASYNC_B128
        __builtin_amdgcn_global_load_async_to_lds_b128(
            (global_v4i_p)(uintptr_t)src,
            (lds_v4i_p)(uintptr_t)&sb[tid * 4],
            /*imm offset=*/0, /*cpol=*/0);
#else
        // Fallback: raw CDNA5 encoding. VDST = 32-bit LDS byte offset,
        // VADDR = 64-bit global address, SADDR = off.
        unsigned lds_off = (unsigned)(uintptr_t)&sb[tid * 4];
        asm volatile("global_load_async_to_lds_b128 %0, %1, off"
                     :: "v"(lds_off), "v"(src) : "memory");
#endif
#if HAS_WAIT_ASYNC
        __builtin_amdgcn_s_wait_asynccnt(0);
#else
        asm volatile("s_wait_asynccnt 0x0" ::: "memory");
#endif
        // Each thread consumes exactly the LDS slots it wrote -> no barrier needed.
    }

    const float4 bb = *reinterpret_cast<const float4*>(&sb[tid * 4]);

    // Token indices are blockIdx-uniform -> compiler scalarizes to s_load.
    int idx[TB];
#pragma unroll
    for (int k = 0; k < TB; ++k) idx[k] = widx[t0 + k];

    // Base of this thread's 4-row window of W; the 4 row strides become
    // 24-bit immediate offsets (8192 floats = 32 KB apart).
    const float* __restrict__ wrow = Wm + (size_t)j * NLOC;

#pragma unroll
    for (int k = 0; k < TB; ++k) {
        const float* p = wrow + idx[k];        // gather column idx[k]
        float4 o;
        o.x = p[0 * NLOC] + bb.x;
        o.y = p[1 * NLOC] + bb.y;
        o.z = p[2 * NLOC] + bb.z;
        o.w = p[3 * NLOC] + bb.w;
        // Perfectly coalesced 512 B/warp float4 store (dominant traffic: 1 GiB).
        *reinterpret_cast<float4*>(out + (size_t)(t0 + k) * NLOC + j) = o;
    }
}

extern "C" void kernel_launch(void* const* d_in, const int* in_sizes, int n_in,
                              void* d_out, int out_size, void* d_ws, size_t ws_size,
                              hipStream_t stream) {
    (void)in_sizes; (void)n_in; (void)out_size; (void)d_ws; (void)ws_size;
    const int*   w = (const int*)  d_in[0];   // indices
    const float* W = (const float*)d_in[1];   // [8192, 8192]
    const float* b = (const float*)d_in[2];   // [8192]
    float* out = (float*)d_out;               // [32768, 8192]

    dim3 grid(TOKENS / TB, NLOC / JTILE, 1);  // (4096, 8): j-slab varies slowest
    fnet_gather_bias_kernel<<<grid, 256, 0, stream>>>(w, W, b, out);
}